// SSM_65859028516938
// MI455X (gfx1250) — compile-verified
//
#include <hip/hip_runtime.h>
#include <hip/hip_bf16.h>
#include <stdint.h>

// Problem dims (from reference)
#define BATCH  2
#define SEQ    2048
#define DMODEL 768
#define DSTATE 16
#define CHUNK  16
#define NCHUNK (SEQ / CHUNK)   // 128

typedef __attribute__((ext_vector_type(2))) float v2f;
typedef __attribute__((ext_vector_type(8))) float v8f;

// Workspace layout, in floats
#define WS_U     0                                   // u:    BATCH*SEQ*DSTATE
#define WS_HLOC  (WS_U    + BATCH*SEQ*DSTATE)        // hloc: BATCH*SEQ*DSTATE
#define WS_P     (WS_HLOC + BATCH*SEQ*DSTATE)        // P:    17*256  (dA^0..dA^16)
#define WS_G     (WS_P    + 17*DSTATE*DSTATE)        // g:    BATCH*NCHUNK*DSTATE
#define WS_CARRY (WS_G    + BATCH*NCHUNK*DSTATE)     // carryIn: BATCH*NCHUNK*DSTATE

// ---------------------------------------------------------------------------
// Pass 1: u[b,s,d] = sum_m B[b,s,m,d] * x[b,s,m]
// One wave per (b,s). Each lane streams contiguous 64B of B per m -> fully
// coalesced 2KB per wave per iteration. HBM-bound (~201MB of B).
// ---------------------------------------------------------------------------
__global__ __launch_bounds__(256) void k_input_proj(const float* __restrict__ x,
                                                    const float* __restrict__ Bm,
                                                    float* __restrict__ u)
{
    const int wid  = blockIdx.x * 8 + (threadIdx.x >> 5);
    const int lane = threadIdx.x & 31;
    const int b = wid / SEQ, s = wid % SEQ;
    const size_t base = ((size_t)b * SEQ + s) * DMODEL;

    float acc[DSTATE];
#pragma unroll
    for (int d = 0; d < DSTATE; ++d) acc[d] = 0.f;

    for (int m = lane; m < DMODEL; m += 32) {
        const float xv = x[base + m];
        const float4* bp = (const float4*)(Bm + (base + m) * DSTATE);
        __builtin_prefetch(bp + 32 * 4, 0, 1);  // next m-iteration's line
        float4 b0 = bp[0], b1 = bp[1], b2 = bp[2], b3 = bp[3];
        acc[0]  += b0.x * xv; acc[1]  += b0.y * xv; acc[2]  += b0.z * xv; acc[3]  += b0.w * xv;
        acc[4]  += b1.x * xv; acc[5]  += b1.y * xv; acc[6]  += b1.z * xv; acc[7]  += b1.w * xv;
        acc[8]  += b2.x * xv; acc[9]  += b2.y * xv; acc[10] += b2.z * xv; acc[11] += b2.w * xv;
        acc[12] += b3.x * xv; acc[13] += b3.y * xv; acc[14] += b3.z * xv; acc[15] += b3.w * xv;
    }

    // wave32 butterfly all-reduce over m-partitions
#pragma unroll
    for (int mask = 16; mask >= 1; mask >>= 1) {
#pragma unroll
        for (int d = 0; d < DSTATE; ++d)
            acc[d] += __shfl_xor(acc[d], mask, 32);
    }

    if (lane == 0) {
        float4* up = (float4*)(u + ((size_t)b * SEQ + s) * DSTATE);
        up[0] = make_float4(acc[0],  acc[1],  acc[2],  acc[3]);
        up[1] = make_float4(acc[4],  acc[5],  acc[6],  acc[7]);
        up[2] = make_float4(acc[8],  acc[9],  acc[10], acc[11]);
        up[3] = make_float4(acc[12], acc[13], acc[14], acc[15]);
    }
}

// ---------------------------------------------------------------------------
// Pass 2: P_k = (exp(A))^k for k=0..16. One block, 256 threads (d,k).
// ---------------------------------------------------------------------------
__global__ __launch_bounds__(256) void k_powers(const float* __restrict__ A,
                                                float* __restrict__ P)
{
    __shared__ float dAs[256], Pa[256], Pb[256];
    const int t = threadIdx.x;
    const int d = t >> 4, k = t & 15;
    dAs[t] = expf(A[t]);
    Pa[t]  = (d == k) ? 1.f : 0.f;
    P[t]   = Pa[t];                       // P_0 = I
    __syncthreads();
    float* cur = Pa;
    float* nxt = Pb;
    for (int s = 1; s <= 16; ++s) {
        float sum = 0.f;
#pragma unroll
        for (int j = 0; j < 16; ++j)
            sum += dAs[d * 16 + j] * cur[j * 16 + k];
        nxt[t] = sum;
        P[s * 256 + t] = sum;
        __syncthreads();
        float* tmp = cur; cur = nxt; nxt = tmp;
    }
}

// ---------------------------------------------------------------------------
// Pass 3: per-chunk local prefix via WMMA.
//   H_local[i,:] = sum_{s=0..15} (Shift_s U)[i,:] @ P_s^T   (16x16x16 matmuls)
// One wave per chunk (256 chunks total). P_s is shared -> true batched matmul.
// The u-tile is staged into LDS with the CDNA5 async global->LDS copy, and is
// zero-padded with 16 rows below so the time-shift never branches (no EXEC
// juggling around the WMMA region; all A/B fetches are plain ds_load_b64s).
// ---------------------------------------------------------------------------
__global__ __launch_bounds__(256) void k_chunk_scan(const float* __restrict__ u,
                                                    const float* __restrict__ P,
                                                    float* __restrict__ hloc,
                                                    float* __restrict__ g)
{
    __shared__ float Ps[16 * 256];   // P_0..P_15
    __shared__ float Us[8][512];     // per wave: rows [-16..-1] = 0, [0..15] = U

    const int tid = threadIdx.x;
    for (int idx = tid; idx < 16 * 256; idx += 256) Ps[idx] = P[idx];

    const int w    = tid >> 5;
    const int lane = tid & 31;
    const int wid  = blockIdx.x * 8 + w;
    const int b = wid / NCHUNK, c = wid % NCHUNK;

    // zero pad rows (first 256 floats): 32 lanes x 2 x float4
    {
        float4* zp = (float4*)&Us[w][0];
        zp[lane]      = make_float4(0.f, 0.f, 0.f, 0.f);
        zp[lane + 32] = make_float4(0.f, 0.f, 0.f, 0.f);
    }

    // Async copy this wave's 1KB u-tile into the upper half (2 x b128 per lane).
    {
        const unsigned tile_bytes = (unsigned)(((b * NCHUNK + c) * 256) * 4);
        const unsigned ldsbase    = (unsigned)(size_t)(&Us[w][256]);
#pragma unroll
        for (int j = 0; j < 2; ++j) {
            const unsigned off = (unsigned)((j * 32 + lane) * 16);
            asm volatile("global_load_async_to_lds_b128 %0, %1, %2"
                         :: "v"(ldsbase + off), "v"(tile_bytes + off), "s"(u)
                         : "memory");
        }
        asm volatile("s_wait_asynccnt 0" ::: "memory");
    }
    __syncthreads();

    const int  i  = lane & 15;       // A-matrix row (time-within-chunk) / B col
    const bool hi = lane >= 16;
    const float* Urows = &Us[w][256];  // row r at Urows[r*16], valid for r in [-16, 16)

    v8f acc = {};
    for (int s = 0; s < 16; ++s) {
        const float* Pm = &Ps[s * 256];
        const int ui = i - s;        // shifted time row; negative -> zero pad
#pragma unroll
        for (int q = 0; q < 4; ++q) {
            const int k0 = q * 4 + (hi ? 2 : 0);
            v2f Av, Bv;
            // A (16x4 tile of Shift_s U): lanes 0-15 hold K={4q,4q+1}, 16-31 K={4q+2,4q+3}
            Av.x = Urows[ui * 16 + k0];
            Av.y = Urows[ui * 16 + k0 + 1];
            // B (4x16 tile of P_s^T): B[kk,n] = P_s[n, 4q+kk]
            Bv.x = Pm[i * 16 + k0];
            Bv.y = Pm[i * 16 + k0 + 1];
            acc = __builtin_amdgcn_wmma_f32_16x16x4_f32(
                false, Av, false, Bv, (short)0, acc, false, false);
        }
    }

    // D layout: VGPR r -> (M = r + (hi?8:0), N = lane&15); M = time, N = state
    const size_t hb = (((size_t)b * SEQ) + (size_t)c * CHUNK) * DSTATE;
#pragma unroll
    for (int r = 0; r < 8; ++r) {
        const int ti = r + (hi ? 8 : 0);
        hloc[hb + (size_t)ti * DSTATE + i] = acc[r];
    }
    if (hi)  // row i=15 (chunk total) lives in VGPR7, lanes 16-31
        g[((b * NCHUNK) + c) * DSTATE + i] = acc[7];
}

// ---------------------------------------------------------------------------
// Pass 4: sequential carry over chunks: carry_{c+1} = P16 * carry_c + g_c.
// 128 tiny steps; single wave (2 batches x 16 states = 32 lanes).
// ---------------------------------------------------------------------------
__global__ __launch_bounds__(32) void k_carry(const float* __restrict__ P,
                                              const float* __restrict__ g,
                                              float* __restrict__ carry)
{
    __shared__ float h[32];
    const int lane = threadIdx.x;
    const int b = lane >> 4, d = lane & 15;
    const float* P16 = P + 16 * 256;
    float row[16];
#pragma unroll
    for (int k = 0; k < 16; ++k) row[k] = P16[d * 16 + k];
    h[lane] = 0.f;
    __syncthreads();
    for (int c = 0; c < NCHUNK; ++c) {
        carry[((b * NCHUNK) + c) * DSTATE + d] = h[lane];  // carry INTO chunk c
        float sum = g[((b * NCHUNK) + c) * DSTATE + d];
#pragma unroll
        for (int k = 0; k < 16; ++k) sum += row[k] * h[b * 16 + k];
        __syncthreads();
        h[lane] = sum;
        __syncthreads();
    }
}

// ---------------------------------------------------------------------------
// Pass 5: fix-up + readout. h[b,s,:] = hloc + P_{i+1} * carryIn; then
//   y[b,s,m] = sum_d h[d]*C[b,s,m,d] + D[m]*x[b,s,m].
// One block per (b,s); lanes stream contiguous 64B of C. HBM-bound (~201MB).
// ---------------------------------------------------------------------------
__global__ __launch_bounds__(256) void k_readout(const float* __restrict__ x,
                                                 const float* __restrict__ C,
                                                 const float* __restrict__ Dv,
                                                 const float* __restrict__ hloc,
                                                 const float* __restrict__ P,
                                                 const float* __restrict__ carry,
                                                 float* __restrict__ y)
{
    __shared__ float hs[DSTATE];
    const int bs = blockIdx.x;
    const int b = bs / SEQ, s = bs % SEQ;
    const int c = s >> 4, i = s & 15;
    const int tid = threadIdx.x;

    if (tid < DSTATE) {
        float v = hloc[(size_t)bs * DSTATE + tid];
        const float* Pi = P + (i + 1) * 256 + tid * 16;
        const float* cr = carry + ((b * NCHUNK) + c) * DSTATE;
#pragma unroll
        for (int k = 0; k < 16; ++k) v += Pi[k] * cr[k];
        hs[tid] = v;
    }
    __syncthreads();

    float hr[16];
#pragma unroll
    for (int k = 0; k < 16; ++k) hr[k] = hs[k];

    const size_t base = (size_t)bs * DMODEL;
#pragma unroll
    for (int mm = 0; mm < DMODEL / 256; ++mm) {
        const int m = tid + mm * 256;
        const float4* cp = (const float4*)(C + (base + m) * DSTATE);
        __builtin_prefetch(cp + 256 * 4, 0, 1);
        float4 c0 = cp[0], c1 = cp[1], c2 = cp[2], c3 = cp[3];
        float acc = c0.x*hr[0]  + c0.y*hr[1]  + c0.z*hr[2]  + c0.w*hr[3]
                  + c1.x*hr[4]  + c1.y*hr[5]  + c1.z*hr[6]  + c1.w*hr[7]
                  + c2.x*hr[8]  + c2.y*hr[9]  + c2.z*hr[10] + c2.w*hr[11]
                  + c3.x*hr[12] + c3.y*hr[13] + c3.z*hr[14] + c3.w*hr[15];
        acc += Dv[m] * x[base + m];
        y[base + m] = acc;
    }
}

// ---------------------------------------------------------------------------
extern "C" void kernel_launch(void* const* d_in, const int* in_sizes, int n_in,
                              void* d_out, int out_size, void* d_ws, size_t ws_size,
                              hipStream_t stream)
{
    const float* x  = (const float*)d_in[0];
    const float* Bm = (const float*)d_in[1];
    const float* C  = (const float*)d_in[2];
    const float* A  = (const float*)d_in[3];
    const float* Dv = (const float*)d_in[4];
    float* y = (float*)d_out;

    float* ws    = (float*)d_ws;
    float* u     = ws + WS_U;
    float* hloc  = ws + WS_HLOC;
    float* P     = ws + WS_P;
    float* g     = ws + WS_G;
    float* carry = ws + WS_CARRY;

    // 1) u = B^T x            (HBM-bound: streams 201MB of B)
    k_input_proj<<<(BATCH * SEQ) / 8, 256, 0, stream>>>(x, Bm, u);
    // 2) matrix powers of exp(A)
    k_powers<<<1, 256, 0, stream>>>(A, P);
    // 3) chunk-local prefix scan on WMMA (256 parallel 16-step chunks)
    k_chunk_scan<<<(BATCH * NCHUNK) / 8, 256, 0, stream>>>(u, P, hloc, g);
    // 4) sequential chunk carries (128 tiny matvec steps)
    k_carry<<<1, 32, 0, stream>>>(P, g, carry);
    // 5) fix-up + readout       (HBM-bound: streams 201MB of C)
    k_readout<<<BATCH * SEQ, 256, 0, stream>>>(x, C, Dv, hloc, P, carry, y);
}